// GoEgate_55525337203004
// MI455X (gfx1250) — compile-verified
//
#include <hip/hip_runtime.h>
#include <hip/hip_bf16.h>

#define DIM    1024
#define DG     256
#define NEXP   16
#define NN     17
#define NTOK   8192
#define LDW    258   // LDS row stride (even -> b64-aligned pairs, conflict-friendly)

// workspace layout (floats)
#define WS_C     0            // 17*256
#define WS_A     4352         // 17 (padded 32)
#define WS_AM    4384         // 17*17
#define WS_WMLP  4736         // packed Wmlp: 1024*256
#define WS_W0    (WS_WMLP + DIM * DG)   // packed W0: 256*256
#define WS_W1    (WS_W0 + DG * DG)      // packed W1: 256*256

typedef __attribute__((ext_vector_type(2))) float v2f;
typedef __attribute__((ext_vector_type(8))) float v8f;

__device__ __forceinline__ v8f wmma4(v2f a, v2f b, v8f c) {
  // D = A(16x4,f32) * B(4x16,f32) + C(16x16,f32)
  return __builtin_amdgcn_wmma_f32_16x16x4_f32(false, a, false, b, (short)0, c,
                                               false, false);
}

// ---------------------------------------------------------------------------
// Pack a row-major [K x 256] weight into WMMA-B-friendly pairs:
//   dst[((k>>1)*256 + n)*2 + (k&1)] = src[k*256 + n]
// so a B-frag (rows kk,kk+1 at col n) is one aligned, coalesced b64 load.
// ---------------------------------------------------------------------------
__global__ __launch_bounds__(256) void k_pack(const float* __restrict__ src,
                                              float* __restrict__ dst, int total)
{
  for (int idx = blockIdx.x * blockDim.x + threadIdx.x; idx < total;
       idx += gridDim.x * blockDim.x) {
    int k = idx >> 8, n = idx & 255;
    dst[(((k >> 1) << 8) + n) * 2 + (k & 1)] = src[idx];
  }
}

// ---------------------------------------------------------------------------
// Setup: normalized adjacency A, shared expert aggregation constants
//   exp  = relu(X @ W_struct)   [16,256]
//   z0e  = exp @ W0             [16,256]
//   c[n] = sum_{m<16} A[n,m] z0e[m]   [17,256];   a[n] = A[n,16]
// ---------------------------------------------------------------------------
__global__ __launch_bounds__(1024) void k_setup(
    const float* __restrict__ X, const float* __restrict__ Wst,
    const float* __restrict__ W0, const float* __restrict__ adj,
    float* __restrict__ cbuf, float* __restrict__ abuf, float* __restrict__ Abuf)
{
  __shared__ float sDinv[NN];
  __shared__ float sA[NN * NN];
  __shared__ float sE[NEXP * DG];
  __shared__ float sZ[NEXP * DG];
  const int tid = threadIdx.x;

  if (tid < NN) {
    float dg = 0.f;
    for (int m = 0; m < NN; ++m) dg += adj[tid * NN + m];
    sDinv[tid] = dg > 0.f ? 1.f / sqrtf(dg) : 0.f;
  }
  __syncthreads();
  for (int idx = tid; idx < NN * NN; idx += blockDim.x) {
    int n = idx / NN, m = idx % NN;
    float v = adj[idx] * sDinv[n] * sDinv[m];
    sA[idx] = v;
    Abuf[idx] = v;
  }
  if (tid < NN) abuf[tid] = adj[tid * NN + NEXP] * sDinv[tid] * sDinv[NEXP];

  for (int idx = tid; idx < NEXP * DG; idx += blockDim.x) {  // exp
    int e = idx >> 8, j = idx & 255;
    float s = 0.f;
    for (int k = 0; k < DIM; ++k) s = fmaf(X[e * DIM + k], Wst[k * DG + j], s);
    sE[idx] = fmaxf(s, 0.f);
  }
  __syncthreads();
  for (int idx = tid; idx < NEXP * DG; idx += blockDim.x) {  // z0e
    int e = idx >> 8, j = idx & 255;
    float s = 0.f;
    for (int k = 0; k < DG; ++k) s = fmaf(sE[e * DG + k], W0[k * DG + j], s);
    sZ[idx] = s;
  }
  __syncthreads();
  for (int idx = tid; idx < NN * DG; idx += blockDim.x) {    // c
    int n = idx >> 8, j = idx & 255;
    float s = 0.f;
    for (int m = 0; m < NEXP; ++m) s = fmaf(sA[n * NN + m], sZ[m * DG + j], s);
    cbuf[idx] = s;
  }
}

// ---------------------------------------------------------------------------
// Fused per-16-token-tile kernel. 16 waves (512 thr): wave w <-> graph node w.
//  1) h = relu(x_tile @ Wmlp)   wave w does cols [16w,16w+16), K=1024 -> LDS
//  2) u = h @ W0                -> LDS
//  3) K-chunked: p[w] = sum_m A[w,m] relu(c[m]+a[m]*u)  (wave-private LDS)
//     then y2_w accum via WMMA:  y2_w = p_w @ W1  (relu deferred)
//  4) scores[t,w] = relu(y2) . Wproj  (in-register + xor-shuffle reduce)
// All B-frags come from packed weights: one b64 load per WMMA.
// ---------------------------------------------------------------------------
__global__ __launch_bounds__(512) void k_main(
    const float* __restrict__ x, const float* __restrict__ WmlpP,
    const float* __restrict__ W0P, const float* __restrict__ W1P,
    const float* __restrict__ Wproj, const float* __restrict__ cbuf,
    const float* __restrict__ abuf, const float* __restrict__ Abuf,
    float* __restrict__ out)
{
  __shared__ float sU[16 * LDW];        // u[t][d]
  __shared__ float sC[NN * LDW];        // c[m][d]
  __shared__ float sHP[16 * 16 * 18];   // phase 1-2: h[t][d]; chunks: p[node][t][kk]

  const int tid  = threadIdx.x;
  const int lane = tid & 31;
  const int w    = tid >> 5;            // wave id == node id (0..15)
  const int mrow = lane & 15;           // M row (A) / N col (B,C)
  const int khi  = (lane >> 4) << 1;    // K sub-offset: 0 or 2
  const int rb   = (lane >> 4) << 3;    // C/D row offset: 0 or 8
  const int t0   = blockIdx.x * 16;

  for (int idx = tid; idx < NN * DG; idx += 512) {
    int m = idx >> 8, d = idx & 255;
    sC[m * LDW + d] = cbuf[idx];
  }
  float arow[NN], avec[NN];
#pragma unroll
  for (int j = 0; j < NN; ++j) { arow[j] = Abuf[w * NN + j]; avec[j] = abuf[j]; }

  v8f zero;
#pragma unroll
  for (int i = 0; i < 8; ++i) zero[i] = 0.f;

  // ---- phase 1: h = relu(x @ Wmlp), K = 1024 ----
  {
    const int n0 = w * 16;
    v8f acc = zero;
    const float* xrow = x + (size_t)(t0 + mrow) * DIM;
    // packed B base for this wave's column slice + this half-wave's k pair
    const float* bb = WmlpP + ((khi >> 1) * DG + n0 + mrow) * 2;
    for (int k0 = 0; k0 < DIM; k0 += 4) {
      const int kk = k0 + khi;
      v2f a = *(const v2f*)(xrow + kk);
      v2f b = *(const v2f*)(bb + (k0 >> 1) * (DG * 2));
      acc = wmma4(a, b, acc);
    }
#pragma unroll
    for (int i = 0; i < 8; ++i)
      sHP[(i + rb) * LDW + n0 + mrow] = fmaxf(acc[i], 0.f);
  }
  __syncthreads();

  // ---- phase 2: u = h @ W0, K = 256 (no relu) ----
  {
    const int n0 = w * 16;
    v8f acc = zero;
    const float* bb = W0P + ((khi >> 1) * DG + n0 + mrow) * 2;
    for (int k0 = 0; k0 < DG; k0 += 4) {
      const int kk = k0 + khi;
      v2f a = *(const v2f*)&sHP[mrow * LDW + kk];
      v2f b = *(const v2f*)(bb + (k0 >> 1) * (DG * 2));
      acc = wmma4(a, b, acc);
    }
#pragma unroll
    for (int i = 0; i < 8; ++i) sU[(i + rb) * LDW + n0 + mrow] = acc[i];
  }
  __syncthreads();

  // ---- phase 3: K-chunked p-build + y2 GEMM (wave-private p => no barriers) ----
  v8f accs[16];
#pragma unroll
  for (int j = 0; j < 16; ++j) accs[j] = zero;

  float* myP = &sHP[w * (16 * 18)];
  const float* bb1 = W1P + ((khi >> 1) * DG + mrow) * 2;  // + (kg/2)*512 + j*32
  for (int c0 = 0; c0 < DG; c0 += 16) {
    if (c0 + 16 < DG) {  // prefetch next chunk of packed W1 (16 rows = 16 KB)
      __builtin_prefetch(W1P + ((c0 + 16) >> 1) * (DG * 2) + ((tid & 127) << 5),
                         0, 1);
    }
    const int d = c0 + mrow;
    float cv[NN];
#pragma unroll
    for (int m = 0; m < NN; ++m) cv[m] = sC[m * LDW + d];
#pragma unroll
    for (int i = 0; i < 8; ++i) {
      const int t = (lane >> 4) + (i << 1);
      const float u = sU[t * LDW + d];
      float s = 0.f;
#pragma unroll
      for (int m = 0; m < NN; ++m) {
        float g = fmaxf(fmaf(avec[m], u, cv[m]), 0.f);
        s = fmaf(arow[m], g, s);
      }
      myP[t * 18 + mrow] = s;
    }
    // 4 k-steps x 16 N-tiles; A-frag reused across all N-tiles
#pragma unroll
    for (int ks = 0; ks < 4; ++ks) {
      const int kk = (ks << 2) + khi;
      v2f a = *(const v2f*)&myP[mrow * 18 + kk];
      const float* bk = bb1 + ((c0 + (ks << 2)) >> 1) * (DG * 2);
#pragma unroll
      for (int j = 0; j < 16; ++j) {
        v2f b = *(const v2f*)(bk + (j << 5));
        accs[j] = wmma4(a, b, accs[j]);
      }
    }
  }

  // ---- phase 4: scores[t, w] = relu(y2) . Wproj ----
  float sc[8];
#pragma unroll
  for (int i = 0; i < 8; ++i) sc[i] = 0.f;
#pragma unroll
  for (int j = 0; j < 16; ++j) {
    const float wp = Wproj[(j << 4) + mrow];
#pragma unroll
    for (int i = 0; i < 8; ++i)
      sc[i] = fmaf(fmaxf(accs[j][i], 0.f), wp, sc[i]);
  }
#pragma unroll
  for (int i = 0; i < 8; ++i) {
    float v = sc[i];
    v += __shfl_xor(v, 1, 32);
    v += __shfl_xor(v, 2, 32);
    v += __shfl_xor(v, 4, 32);
    v += __shfl_xor(v, 8, 32);
    sc[i] = v;
  }
  if (mrow == 0) {
#pragma unroll
    for (int i = 0; i < 8; ++i)
      out[(size_t)(t0 + rb + i) * NEXP + w] = sc[i];
  }
}

extern "C" void kernel_launch(void* const* d_in, const int* in_sizes, int n_in,
                              void* d_out, int out_size, void* d_ws, size_t ws_size,
                              hipStream_t stream) {
  const float* x    = (const float*)d_in[0];   // [8192,1024]
  const float* X    = (const float*)d_in[1];   // [16,1024]
  const float* Wmlp = (const float*)d_in[2];   // [1024,256]
  const float* Wst  = (const float*)d_in[3];   // [1024,256]
  const float* Wc   = (const float*)d_in[4];   // [2,256,256]
  const float* Wpr  = (const float*)d_in[5];   // [256,1]
  const float* adj  = (const float*)d_in[6];   // [17,17]
  float* ws    = (float*)d_ws;
  float* cbuf  = ws + WS_C;
  float* abuf  = ws + WS_A;
  float* Abuf  = ws + WS_AM;
  float* WmlpP = ws + WS_WMLP;
  float* W0P   = ws + WS_W0;
  float* W1P   = ws + WS_W1;
  float* out   = (float*)d_out;                // [8192,16]

  k_pack<<<256, 256, 0, stream>>>(Wmlp, WmlpP, DIM * DG);
  k_pack<<<64, 256, 0, stream>>>(Wc, W0P, DG * DG);
  k_pack<<<64, 256, 0, stream>>>(Wc + DG * DG, W1P, DG * DG);
  k_setup<<<1, 1024, 0, stream>>>(X, Wst, Wc, adj, cbuf, abuf, Abuf);
  k_main<<<NTOK / 16, 512, 0, stream>>>(x, WmlpP, W0P, W1P, Wpr, cbuf, abuf,
                                        Abuf, out);
}